// HebbianConv2d_89111981457956
// MI455X (gfx1250) — compile-verified
//
#include <hip/hip_runtime.h>

typedef __attribute__((ext_vector_type(2))) float    v2f;
typedef __attribute__((ext_vector_type(8))) float    v8f;
typedef __attribute__((ext_vector_type(4))) unsigned v4u;
typedef __attribute__((ext_vector_type(8))) unsigned v8u;

#define BB   32
#define CIN  96
#define EXC  102
#define INH  26
#define HW   4096            // 64*64
#define NPIX (BB * HW)       // 131072

#define EXCP 104             // EXC padded to multiple of 4
#define INHP 28              // INH padded to multiple of 4

static __device__ __forceinline__ v8f wmma4(v2f a, v2f b, v8f c) {
  return __builtin_amdgcn_wmma_f32_16x16x4_f32(false, a, false, b, (short)0, c,
                                               false, false);
}

// ---- force-global (AS1) memory helpers: guarantees global_load/global_store
typedef const float __attribute__((address_space(1)))* gcptr;
typedef float __attribute__((address_space(1)))*       gptr;
typedef const v2f __attribute__((address_space(1)))*   g2cptr;

static __device__ __forceinline__ float gload1(const float* p) {
  return *(gcptr)(unsigned long long)(uintptr_t)p;
}
static __device__ __forceinline__ v2f gload2(const float* p) {
  return *(g2cptr)(unsigned long long)(uintptr_t)p;
}
static __device__ __forceinline__ void gstore1(float* p, float v) {
  *(gptr)(unsigned long long)(uintptr_t)p = v;
}

// ---- CDNA5 per-lane async global->LDS copy (ASYNCcnt path).
// LDS operand = low 32 bits of the generic pointer (ISA 10.2: LDS aperture
// addresses truncate to the LDS byte offset).
static __device__ __forceinline__ void async_copy_b32(void* lds_dst,
                                                      const float* gsrc) {
  const unsigned lds_off = (unsigned)(uintptr_t)lds_dst;
  const unsigned long long ga = (unsigned long long)(uintptr_t)gsrc;
  asm volatile("global_load_async_to_lds_b32 %0, %1, off"
               :: "v"(lds_off), "v"(ga) : "memory");
}
static __device__ __forceinline__ void wait_async0() {
#if __has_builtin(__builtin_amdgcn_s_wait_asynccnt)
  __builtin_amdgcn_s_wait_asynccnt(0);
#else
  asm volatile("s_wait_asynccnt 0x0" ::: "memory");
#endif
}

// ---- CDNA5 Tensor Data Mover: 1-D tile DMA global->LDS (TENSORcnt path).
// D# per ISA 8.3/8.4: group0 = {count=1, lds_addr, global_addr, type=2},
// group1 = {data_size=4B, tensor_dim0 = tile_dim0 = nElems, stride = nElems}.
// VADDR2/VADDR3 omitted (<=2D tensor). Issue from one wave per block.
static __device__ __forceinline__ void tdm_load_1d(void* lds_dst,
                                                   const float* gsrc,
                                                   unsigned nElems) {
  const unsigned long long ga = (unsigned long long)(uintptr_t)gsrc;
  const unsigned lds = (unsigned)(uintptr_t)lds_dst;
  v4u g0;
  g0.x = 1u;                                             // count=1 (user D#)
  g0.y = lds;                                            // lds_addr (bytes)
  g0.z = (unsigned)ga;                                   // global_addr[31:0]
  g0.w = (unsigned)((ga >> 32) & 0x01FFFFFFu) | (2u << 30); // addr[56:32]|type=2
  v8u g1;
  g1.s0 = 2u << 16;                                      // data_size=4B
  g1.s1 = (nElems & 0xFFFFu) << 16;                      // tensor_dim0[15:0]
  g1.s2 = (nElems >> 16) & 0xFFFFu;                      // tensor_dim0[31:16]
  g1.s3 = (nElems & 0xFFFFu) << 16;                      // tile_dim0
  g1.s4 = 0u;                                            // tile_dim1/2 unused
  g1.s5 = nElems;                                        // dim0_stride[31:0]
  g1.s6 = 0u;
  g1.s7 = 0u;
  asm volatile("tensor_load_to_lds %0, %1" :: "s"(g0), "s"(g1) : "memory");
}
static __device__ __forceinline__ void wait_tensor_le(int n) {
#if __has_builtin(__builtin_amdgcn_s_wait_tensorcnt)
  if (n == 0) __builtin_amdgcn_s_wait_tensorcnt(0);
  else        __builtin_amdgcn_s_wait_tensorcnt(1);
#else
  if (n == 0) asm volatile("s_wait_tensorcnt 0x0" ::: "memory");
  else        asm volatile("s_wait_tensorcnt 0x1" ::: "memory");
#endif
}

// ---------------------------------------------------------------------------
// Layout repacks (one streaming pass each; trivial vs the conv FLOPs).
// ---------------------------------------------------------------------------
__global__ void repack_x(const float* __restrict__ x, float* __restrict__ xh) {
  const int idx = blockIdx.x * blockDim.x + threadIdx.x;   // over NPIX*CIN
  if (idx >= NPIX * CIN) return;
  const int c = idx % CIN;
  const int P = idx / CIN;
  const int b = P >> 12, hw = P & 4095;
  xh[idx] = x[((size_t)b * CIN + c) * HW + hw];
}

// w[Cout,Cin,5,5] -> wT[25][WPAD][CINP], zero-padded (oc>=Cout or ci>=Cin -> 0)
template<int Cout, int Cin, int WPAD, int CINP>
__global__ void repack_w(const float* __restrict__ w, float* __restrict__ wT) {
  const int idx = blockIdx.x * blockDim.x + threadIdx.x;
  if (idx >= 25 * WPAD * CINP) return;
  const int ci  = idx % CINP;
  const int rem = idx / CINP;
  const int oc  = rem % WPAD;
  const int tap = rem / WPAD;
  float v = 0.f;
  if (oc < Cout && ci < Cin) v = w[((size_t)oc * Cin + ci) * 25 + tap];
  wT[idx] = v;
}

// ---------------------------------------------------------------------------
// 5x5 reflect-pad conv, NHWC, implicit GEMM on the f32 WMMA pipe.
// Block = 128 threads (4 waves). Each wave: 2 M-tiles x 2 N-tiles (4 accs),
// so each K-step is 4 WMMAs fed by 2 global b64 (A) + 2 LDS b64 (B).
// Weight tiles (32 oc x CINP) are double-buffered in LDS via the Tensor Data
// Mover: wave 0 issues tap t+1's DMA, then s_wait_tensorcnt releases tap t
// while the prefetch stays in flight.
// MODE: 0 = identity, 1 = relu, 2 = add `addsrc` before store.
// ---------------------------------------------------------------------------
template<int CINP, int COUTP, int WPAD, int MODE>
__global__ __launch_bounds__(128) void conv5_wmma(
    const float* __restrict__ in, const float* __restrict__ wT,
    const float* __restrict__ addsrc, float* __restrict__ out)
{
  const int tid  = threadIdx.x;
  const int lane = tid & 31;
  const int wave = tid >> 5;
  const int row  = lane & 15;
  const int koff = (lane >> 4) << 1;        // K sub-offset per ISA A/B layout

  constexpr int tileElems = 32 * CINP;
  __shared__ float wlds[2][tileElems];      // TDM double buffer

  // Two M-tiles (16 pixels each) per wave
  const int mt0 = blockIdx.x * 8 + wave * 2;
  int b_[2], h_[2], w_[2];
  #pragma unroll
  for (int i = 0; i < 2; ++i) {
    const int m = (mt0 + i) * 16 + row;
    b_[i] = m >> 12;
    h_[i] = (m >> 6) & 63;
    w_[i] = m & 63;
  }
  const int ocblk = blockIdx.y * 32;
  const float* wbase = wT + (size_t)ocblk * CINP;   // + tap*WPAD*CINP per tap

  v8f acc[2][2] = {};

  if (wave == 0)  // prologue: DMA tap 0 into buffer 0
    tdm_load_1d(&wlds[0][0], wbase, tileElems);

  for (int tap = 0; tap < 25; ++tap) {
    const int kh = tap / 5, kw = tap % 5;

    if (wave == 0) {
      if (tap < 24) {   // prefetch next tap into the other buffer, release cur
        tdm_load_1d(&wlds[(tap + 1) & 1][0],
                    wbase + (size_t)(tap + 1) * WPAD * CINP, tileElems);
        wait_tensor_le(1);
      } else {
        wait_tensor_le(0);
      }
    }
    __syncthreads();                        // tap's tile visible to all waves
    const float* wcur = &wlds[tap & 1][0];

    // A pointers: reflect-padded pixel, channel-contiguous (NHWC)
    const float* pa0;
    const float* pa1;
    {
      int rh = h_[0] + kh - 2; rh = (rh < 0) ? -rh : ((rh > 63) ? 126 - rh : rh);
      int rw = w_[0] + kw - 2; rw = (rw < 0) ? -rw : ((rw > 63) ? 126 - rw : rw);
      pa0 = in + (size_t)(b_[0] * HW + rh * 64 + rw) * CINP + koff;
      rh = h_[1] + kh - 2; rh = (rh < 0) ? -rh : ((rh > 63) ? 126 - rh : rh);
      rw = w_[1] + kw - 2; rw = (rw < 0) ? -rw : ((rw > 63) ? 126 - rw : rw);
      pa1 = in + (size_t)(b_[1] * HW + rh * 64 + rw) * CINP + koff;
    }
    const float* pb0 = wcur + (size_t)(0 * 16 + row) * CINP + koff;
    const float* pb1 = wcur + (size_t)(1 * 16 + row) * CINP + koff;

    #pragma unroll 4
    for (int t = 0; t < CINP / 4; ++t) {
      const v2f a0 = gload2(pa0);
      const v2f a1 = gload2(pa1);
      const v2f b0 = *(const v2f*)pb0;    // LDS: ds_load_b64
      const v2f b1 = *(const v2f*)pb1;
      acc[0][0] = wmma4(a0, b0, acc[0][0]);
      acc[0][1] = wmma4(a0, b1, acc[0][1]);
      acc[1][0] = wmma4(a1, b0, acc[1][0]);
      acc[1][1] = wmma4(a1, b1, acc[1][1]);
      pa0 += 4; pa1 += 4; pb0 += 4; pb1 += 4;
    }
    __syncthreads();    // all waves done with this buffer before next overwrite
  }

  // C/D layout: vgpr r holds M = r (+8 for lanes 16..31), N = lane&15
  #pragma unroll
  for (int j = 0; j < 2; ++j) {
    const int oc = ocblk + j * 16 + (lane & 15);
    if (oc < COUTP) {           // padded channels [COUT,COUTP) store exact 0
      #pragma unroll
      for (int i = 0; i < 2; ++i) {
        const int mbase = (mt0 + i) * 16 + ((lane >> 4) << 3);
        #pragma unroll
        for (int r = 0; r < 8; ++r) {
          const int P = mbase + r;                  // global pixel id
          float v = acc[i][j][r];
          if (MODE == 1) v = fmaxf(v, 0.f);
          float* op = out + (size_t)P * COUTP + oc;
          if (MODE == 2) v += gload1(addsrc + (size_t)P * COUTP + oc);
          gstore1(op, v);
        }
      }
    }
  }
}

// ---------------------------------------------------------------------------
// Correlation GEMM: C[ar,bc] = sum_k A[k,ar] * B[k,bc], K = NPIX, split-K
// over grid.z with atomic accumulation. B is NHWC [NPIX,CBP]. A is either
// NHWC [NPIX,CAP] (ANHWC=true) or NCHW [B,CA,HW] (final y in d_out).
// ---------------------------------------------------------------------------
template<int CA, int CAP, int CB, int CBP, bool ANHWC>
__global__ __launch_bounds__(32) void corr_wmma(
    const float* __restrict__ Am, const float* __restrict__ Bm,
    float* __restrict__ C)
{
  const int lane  = threadIdx.x & 31;
  const int row   = lane & 15;
  const int koff  = (lane >> 4) << 1;
  const int atile = blockIdx.x, btile = blockIdx.y;

  const int ar = atile * 16 + row;
  const int bc = btile * 16 + row;
  const bool av = ar < CA, bv = bc < CB;
  const int arc = av ? ar : 0;
  const int bcc = bv ? bc : 0;

  v8f acc = {};
  const int kbase0 = blockIdx.z * 1024;
  for (int k4 = 0; k4 < 1024; k4 += 4) {
    const int k = kbase0 + k4;            // 4-aligned, never straddles a plane
    v2f a, bf;
    if (ANHWC) {
      const float* pa = Am + (size_t)(k + koff) * CAP + arc;
      a.x = av ? gload1(pa)       : 0.f;
      a.y = av ? gload1(pa + CAP) : 0.f;
    } else {
      const int b = k >> 12;
      const float* pa = Am + ((size_t)b * CA + arc) * HW + (k & 4095) + koff;
      if (av) { const v2f t = gload2(pa); a = t; }
      else    { a.x = 0.f; a.y = 0.f; }
    }
    const float* pb = Bm + (size_t)(k + koff) * CBP + bcc;
    bf.x = bv ? gload1(pb)       : 0.f;
    bf.y = bv ? gload1(pb + CBP) : 0.f;
    acc = wmma4(a, bf, acc);
  }

  const int cc = btile * 16 + (lane & 15);
  if (cc < CB) {
    #pragma unroll
    for (int r = 0; r < 8; ++r) {
      const int rr = atile * 16 + r + ((lane >> 4) << 3);
      if (rr < CA) atomicAdd(&C[rr * CB + cc], acc[r]);
    }
  }
}

// ---------------------------------------------------------------------------
// Winner-take-all over channels per pixel (NHWC: channel scan is contiguous).
// ---------------------------------------------------------------------------
template<int CH, int CHP, bool ABSARG>
__global__ void wta_kernel(const float* __restrict__ yin,
                           float* __restrict__ ywta, float* __restrict__ yu)
{
  const int p = blockIdx.x * blockDim.x + threadIdx.x;
  if (p >= NPIX) return;
  const float* base = yin + (size_t)p * CHP;
  int win = 0;
  float bestkey = ABSARG ? fabsf(base[0]) : base[0];
  float bestval = base[0];
  for (int e = 1; e < CH; ++e) {
    const float v   = base[e];
    const float key = ABSARG ? fabsf(v) : v;
    if (key > bestkey) { bestkey = key; bestval = v; win = e; }  // first-max ties
  }
  float* ob = ywta + (size_t)p * CHP;
  for (int e = 0; e < CHP; ++e) ob[e] = 0.f;                     // incl. padding
  ob[win] = bestval;
  atomicAdd(&yu[win], ABSARG ? fabsf(bestval) : bestval);
}

// ---------------------------------------------------------------------------
// Depthwise 5x5 surround-modulation conv, zero padding.
// The 25 coefficients are staged into LDS once per block with a single
// per-lane GLOBAL_LOAD_ASYNC_TO_LDS_B32 (25 active lanes).
// Reads NHWC-padded y_sum, writes final y in NCHW (reference output layout).
// ---------------------------------------------------------------------------
__global__ void sm_conv_kernel(const float* __restrict__ ysum,
                               const float* __restrict__ sm,
                               float* __restrict__ ynchw)
{
  __shared__ float smlds[25];
  if (threadIdx.x < 25) async_copy_b32(&smlds[threadIdx.x], sm + threadIdx.x);
  wait_async0();
  __syncthreads();

  const int idx = blockIdx.x * blockDim.x + threadIdx.x;   // over NPIX*EXC
  if (idx >= NPIX * EXC) return;
  const int e = idx % EXC;
  const int P = idx / EXC;
  const int b = P >> 12;
  const int h = (P >> 6) & 63, w = P & 63;
  float s = 0.f;
  for (int kh = 0; kh < 5; ++kh) {
    const int rh = h + kh - 2;
    if ((unsigned)rh >= 64u) continue;
    for (int kw = 0; kw < 5; ++kw) {
      const int rw = w + kw - 2;
      if ((unsigned)rw >= 64u) continue;
      s += smlds[kh * 5 + kw] * ysum[(size_t)(b * HW + rh * 64 + rw) * EXCP + e];
    }
  }
  ynchw[((size_t)b * EXC + e) * HW + (P & 4095)] = s;
}

// ---------------------------------------------------------------------------
// Hebbian update: upd = yx - yu[row]*w ; normalize by global absmax.
// absmax via atomicMax on float bits (valid for non-negative floats).
// ---------------------------------------------------------------------------
__global__ void upd_pass1(const float* __restrict__ yx, const float* __restrict__ yu,
                          const float* __restrict__ wgt, float* __restrict__ outp,
                          unsigned* __restrict__ amax, int R, int Cc)
{
  const int idx = blockIdx.x * blockDim.x + threadIdx.x;
  if (idx >= R * Cc * 25) return;
  const int rc = idx / 25;
  const int c  = rc % Cc;
  const int r  = rc / Cc;
  const float v = yx[r * Cc + c] - yu[r] * wgt[idx];
  outp[idx] = v;
  atomicMax(amax, __float_as_uint(fabsf(v)));
}

__global__ void upd_pass2(float* __restrict__ outp,
                          const unsigned* __restrict__ amax, int total)
{
  const int idx = blockIdx.x * blockDim.x + threadIdx.x;
  if (idx >= total) return;
  outp[idx] = outp[idx] / (__uint_as_float(*amax) + 1e-8f);
}

// ---------------------------------------------------------------------------
extern "C" void kernel_launch(void* const* d_in, const int* in_sizes, int n_in,
                              void* d_out, int out_size, void* d_ws, size_t ws_size,
                              hipStream_t stream)
{
  (void)in_sizes; (void)n_in; (void)out_size; (void)ws_size;
  const float* x    = (const float*)d_in[0];
  const float* w_ee = (const float*)d_in[1];
  const float* w_ei = (const float*)d_in[2];
  const float* w_ie = (const float*)d_in[3];
  const float* smk  = (const float*)d_in[4];

  float* out = (float*)d_out;
  float* ws  = (float*)d_ws;

  const size_t PLANE_XH = (size_t)NPIX * CIN;    // 12,582,912
  const size_t PLANE_E  = (size_t)NPIX * EXCP;   // 13,631,488
  const size_t PLANE_I  = (size_t)NPIX * INHP;   //  3,670,016

  float* xh     = ws;                            // x in NHWC
  float* y_e    = xh     + PLANE_XH;             // NHWC padded [NPIX,104]
  float* y_sum  = y_e    + PLANE_E;
  float* y_i    = y_sum  + PLANE_E;              // NHWC padded [NPIX,28]
  float* ywta_e = y_i    + PLANE_I;
  float* ywta_i = ywta_e + PLANE_E;
  float* wT_ee  = ywta_i + PLANE_I;              // [25][128][96]
  float* wT_ei  = wT_ee  + 25 * 128 * 96;        // [25][32][104]
  float* wT_ie  = wT_ei  + 25 * 32 * 104;        // [25][128][28]
  float* acc0   = wT_ie  + 25 * 128 * 28;
  float* yx_ee  = acc0;                          // [EXC, CIN]
  float* yx_ei  = yx_ee + EXC * CIN;             // [INH, EXC]
  float* yx_ie  = yx_ei + INH * EXC;             // [EXC, INH]
  float* yu_e   = yx_ie + EXC * INH;
  float* yu_i   = yu_e  + EXC;
  unsigned* amax = (unsigned*)(yu_i + INH);      // 3 scalars

  const size_t accBytes =
      (size_t)(EXC * CIN + INH * EXC + EXC * INH + EXC + INH + 3) * sizeof(float);
  hipMemsetAsync(acc0, 0, accBytes, stream);

  float* y      = out;                           // [B,EXC,H,W] (NCHW)
  float* upd_ee = out + (size_t)NPIX * EXC;      // [EXC,CIN,5,5]
  float* upd_ei = upd_ee + EXC * CIN * 25;       // [INH,EXC,5,5]
  float* upd_ie = upd_ei + INH * EXC * 25;       // [EXC,INH,5,5]

  // Layout repacks
  repack_x<<<(NPIX * CIN + 255) / 256, 256, 0, stream>>>(x, xh);
  repack_w<EXC, CIN, 128,  96><<<(25*128* 96 + 255) / 256, 256, 0, stream>>>(w_ee, wT_ee);
  repack_w<INH, EXC,  32, 104><<<(25* 32*104 + 255) / 256, 256, 0, stream>>>(w_ei, wT_ei);
  repack_w<EXC, INH, 128,  28><<<(25*128* 28 + 255) / 256, 256, 0, stream>>>(w_ie, wT_ie);

  // Convs: grid.x = NPIX/128 pixel-blocks, grid.y = out-channel blocks of 32
  conv5_wmma< 96, EXCP, 128, 0><<<dim3(1024, 4), 128, 0, stream>>>(xh,  wT_ee, nullptr, y_e);
  conv5_wmma<104, INHP,  32, 1><<<dim3(1024, 1), 128, 0, stream>>>(y_e, wT_ei, nullptr, y_i);
  conv5_wmma< 28, EXCP, 128, 2><<<dim3(1024, 4), 128, 0, stream>>>(y_i, wT_ie, y_e, y_sum);

  sm_conv_kernel<<<(NPIX * EXC + 255) / 256, 256, 0, stream>>>(y_sum, smk, y);

  wta_kernel<EXC, EXCP, false><<<(NPIX + 255) / 256, 256, 0, stream>>>(y_e, ywta_e, yu_e);
  wta_kernel<INH, INHP, true ><<<(NPIX + 255) / 256, 256, 0, stream>>>(y_i, ywta_i, yu_i);

  // yx_ee[e,c]=sum ywta_e*x ; yx_ei[i,e]=sum ywta_i*ywta_e ; yx_ie[e,i]=sum y*ywta_i
  corr_wmma<EXC, EXCP, CIN,  CIN, true ><<<dim3(7, 6, 128), 32, 0, stream>>>(ywta_e, xh,     yx_ee);
  corr_wmma<INH, INHP, EXC, EXCP, true ><<<dim3(2, 7, 128), 32, 0, stream>>>(ywta_i, ywta_e, yx_ei);
  corr_wmma<EXC, EXC,  INH, INHP, false><<<dim3(7, 2, 128), 32, 0, stream>>>(y,      ywta_i, yx_ie);

  upd_pass1<<<(EXC*CIN*25 + 255) / 256, 256, 0, stream>>>(yx_ee, yu_e, w_ee, upd_ee, amax + 0, EXC, CIN);
  upd_pass1<<<(INH*EXC*25 + 255) / 256, 256, 0, stream>>>(yx_ei, yu_i, w_ei, upd_ei, amax + 1, INH, EXC);
  upd_pass1<<<(EXC*INH*25 + 255) / 256, 256, 0, stream>>>(yx_ie, yu_e, w_ie, upd_ie, amax + 2, EXC, INH);

  upd_pass2<<<(EXC*CIN*25 + 255) / 256, 256, 0, stream>>>(upd_ee, amax + 0, EXC*CIN*25);
  upd_pass2<<<(INH*EXC*25 + 255) / 256, 256, 0, stream>>>(upd_ei, amax + 1, INH*EXC*25);
  upd_pass2<<<(EXC*INH*25 + 255) / 256, 256, 0, stream>>>(upd_ie, amax + 2, EXC*INH*25);
}